// GIN_38714835206730
// MI455X (gfx1250) — compile-verified
//
#include <hip/hip_runtime.h>

// Problem constants (match reference)
#define N_NODES 50000
#define N_EDGES 800000
#define DIM 128
#define TS 132          // padded LDS row stride in floats (conflict-free WMMA fragment loads)

typedef __attribute__((ext_vector_type(2))) float v2f;
typedef __attribute__((ext_vector_type(8))) float v8f;

// ---------------------------------------------------------------------------
// Zero a float buffer (graph-capture-safe memset replacement)
// ---------------------------------------------------------------------------
__global__ void zero_f32(float* __restrict__ p, int n) {
  int i = blockIdx.x * blockDim.x + threadIdx.x;
  int stride = gridDim.x * blockDim.x;
  for (; i < n; i += stride) p[i] = 0.0f;
}

// ---------------------------------------------------------------------------
// neigh[dst[e]] += h[src[e]]  (segment_sum). One wave32 per edge, one lane
// per 4 consecutive dims: float4 gather + 4 hardware f32 atomics into L2.
// h (25.6MB) is L2-resident on MI455X (192MB L2), so this is L2-atomic bound.
// ---------------------------------------------------------------------------
__global__ __launch_bounds__(256) void scatter_add(
    const float* __restrict__ h,
    const long long* __restrict__ src,
    const long long* __restrict__ dst,
    float* __restrict__ neigh) {
  const int wave = threadIdx.x >> 5;
  const int lane = threadIdx.x & 31;
  const long long e = (long long)blockIdx.x * 8 + wave;
  if (e >= N_EDGES) return;
  const long long s = src[e];
  const long long d = dst[e];
  const float4 v = *(const float4*)(h + s * DIM + lane * 4);
  float* o = neigh + d * DIM + lane * 4;
  unsafeAtomicAdd(o + 0, v.x);   // -> global_atomic_add_f32
  unsafeAtomicAdd(o + 1, v.y);
  unsafeAtomicAdd(o + 2, v.z);
  unsafeAtomicAdd(o + 3, v.w);
}

// ---------------------------------------------------------------------------
// Software-pipelined 16x128 strip GEMM on the fp32 matrix pipe.
// Each wave computes rows [rowb, rowb+16) x all 128 cols with 8 v8f
// accumulators. For each k-step we prefetch the NEXT step's A fragment and
// all 8 B fragments into distinct registers before issuing the current
// step's 8 v_wmma_f32_16x16x4_f32, so the ds_load latency is hidden behind
// WMMA issue instead of an s_wait_dscnt 0 per WMMA pair (only 2 waves/SIMD
// at this LDS footprint, so ILP must hide the latency).
// ---------------------------------------------------------------------------
__device__ __forceinline__ void wmma_strip_gemm(
    const float* __restrict__ T,      // LDS activations, stride TS
    const float* __restrict__ WT,     // LDS weights transposed: WT[n*TS + k]
    int rowb, int m, int hi, v8f acc[8]) {
  const float* arow  = &T[(rowb + m) * TS + 2 * hi];   // A: row m, K split by lane half
  const float* bbase = &WT[m * TS + 2 * hi];           // B: col m of each 16-col tile

  v2f a = *(const v2f*)(arow);
  v2f b[8];
#pragma unroll
  for (int nt = 0; nt < 8; ++nt) b[nt] = *(const v2f*)(bbase + nt * 16 * TS);

#pragma unroll 2
  for (int kk = 0; kk < 32; ++kk) {
    const int k1 = (kk < 31) ? (kk + 1) * 4 : 0;       // wrap: harmless reload
    // prefetch next k-step fragments (9 ds loads in flight)
    const v2f a_n = *(const v2f*)(arow + k1);
    v2f b_n[8];
#pragma unroll
    for (int nt = 0; nt < 8; ++nt) b_n[nt] = *(const v2f*)(bbase + nt * 16 * TS + k1);
    // consume current fragments
#pragma unroll
    for (int nt = 0; nt < 8; ++nt) {
      acc[nt] = __builtin_amdgcn_wmma_f32_16x16x4_f32(
          false, a, false, b[nt], (short)0, acc[nt], false, false);
    }
    a = a_n;
#pragma unroll
    for (int nt = 0; nt < 8; ++nt) b[nt] = b_n[nt];    // renamed away by unroll-2
  }
}

// ---------------------------------------------------------------------------
// Fused GIN layer for a 128-node tile:
//   T   = h + neigh
//   U   = ReLU(T @ W1 + b1)
//   out = U @ W2 + b2 (+ h if residual)
// 256 threads = 8 waves; wave w owns node rows [w*16, w*16+16) of the tile.
// W1^T, W2^T and T all live in LDS (~198KB -> needs CDNA5's 320KB WGP LDS).
// Weights are stored transposed with stride TS=132 so that each WMMA B
// fragment is a single 8B-aligned ds_load_b64 hitting 64 distinct banks.
// ---------------------------------------------------------------------------
__global__ __launch_bounds__(256) void gin_layer(
    const float* __restrict__ h, const float* __restrict__ neigh,
    const float* __restrict__ w1, const float* __restrict__ b1,
    const float* __restrict__ w2, const float* __restrict__ b2,
    float* __restrict__ out, int residual) {
  extern __shared__ float smem[];
  float* LW1 = smem;                 // DIM*TS floats  (W1 transposed)
  float* LW2 = smem + DIM * TS;      // DIM*TS floats  (W2 transposed)
  float* T   = smem + 2 * DIM * TS;  // DIM*TS floats  (activation tile)

  const int tid = threadIdx.x;
  const int r0 = blockIdx.x * 128;

  // --- cooperative load: weights (transposed into LDS) ---
#pragma unroll
  for (int i = 0; i < 16; ++i) {
    const int slot = tid + i * 256;           // 4096 float4 slots
    const int k = slot >> 5;
    const int n4 = (slot & 31) << 2;
    const float4 a = *(const float4*)&w1[k * DIM + n4];
    LW1[(n4 + 0) * TS + k] = a.x;
    LW1[(n4 + 1) * TS + k] = a.y;
    LW1[(n4 + 2) * TS + k] = a.z;
    LW1[(n4 + 3) * TS + k] = a.w;
    const float4 c = *(const float4*)&w2[k * DIM + n4];
    LW2[(n4 + 0) * TS + k] = c.x;
    LW2[(n4 + 1) * TS + k] = c.y;
    LW2[(n4 + 2) * TS + k] = c.z;
    LW2[(n4 + 3) * TS + k] = c.w;
  }
  // --- cooperative load: T = h + neigh (rows beyond N zero-padded) ---
#pragma unroll
  for (int i = 0; i < 16; ++i) {
    const int slot = tid + i * 256;
    const int r = slot >> 5;
    const int c4 = (slot & 31) << 2;
    const int g = r0 + r;
    float4 t = make_float4(0.f, 0.f, 0.f, 0.f);
    if (g < N_NODES) {
      const float4 a = *(const float4*)&h[(long long)g * DIM + c4];
      const float4 b = *(const float4*)&neigh[(long long)g * DIM + c4];
      t = make_float4(a.x + b.x, a.y + b.y, a.z + b.z, a.w + b.w);
    }
    *(float4*)&T[r * TS + c4] = t;
  }
  __syncthreads();

  const int wave = tid >> 5;
  const int lane = tid & 31;
  const int m = lane & 15;
  const int hi = lane >> 4;
  const int rowb = wave * 16;

  // --- GEMM1: T @ W1 ---
  v8f acc[8];
#pragma unroll
  for (int nt = 0; nt < 8; ++nt) acc[nt] = (v8f){};
  wmma_strip_gemm(T, LW1, rowb, m, hi, acc);

  // --- bias + ReLU, write U back into this wave's private strip of T.
  // No __syncthreads needed: only this wave reads/writes rows [rowb,rowb+16),
  // and LDS ops within a wave are processed in order.
#pragma unroll
  for (int nt = 0; nt < 8; ++nt) {
    const int c = nt * 16 + m;
    const float bias = b1[c];
#pragma unroll
    for (int v = 0; v < 8; ++v) {
      const float u = acc[nt][v] + bias;
      T[(rowb + v + 8 * hi) * TS + c] = u > 0.0f ? u : 0.0f;
    }
  }

  // --- GEMM2: U @ W2 ---
  v8f acc2[8];
#pragma unroll
  for (int nt = 0; nt < 8; ++nt) acc2[nt] = (v8f){};
  wmma_strip_gemm(T, LW2, rowb, m, hi, acc2);

  // --- bias (+ residual) -> global ---
#pragma unroll
  for (int nt = 0; nt < 8; ++nt) {
    const int c = nt * 16 + m;
    const float bias = b2[c];
#pragma unroll
    for (int v = 0; v < 8; ++v) {
      const int g = r0 + rowb + v + 8 * hi;
      if (g < N_NODES) {
        float o = acc2[nt][v] + bias;
        if (residual) o += h[(long long)g * DIM + c];
        out[(long long)g * DIM + c] = o;
      }
    }
  }
}

// ---------------------------------------------------------------------------
// Head: out = h @ wh + bh  (single WMMA GEMM, same tiling)
// ---------------------------------------------------------------------------
__global__ __launch_bounds__(256) void head_kernel(
    const float* __restrict__ h,
    const float* __restrict__ wh, const float* __restrict__ bh,
    float* __restrict__ out) {
  extern __shared__ float smem[];
  float* LW = smem;              // DIM*TS  (Wh transposed)
  float* T  = smem + DIM * TS;   // DIM*TS

  const int tid = threadIdx.x;
  const int r0 = blockIdx.x * 128;

#pragma unroll
  for (int i = 0; i < 16; ++i) {
    const int slot = tid + i * 256;
    const int k = slot >> 5;
    const int n4 = (slot & 31) << 2;
    const float4 a = *(const float4*)&wh[k * DIM + n4];
    LW[(n4 + 0) * TS + k] = a.x;
    LW[(n4 + 1) * TS + k] = a.y;
    LW[(n4 + 2) * TS + k] = a.z;
    LW[(n4 + 3) * TS + k] = a.w;
  }
#pragma unroll
  for (int i = 0; i < 16; ++i) {
    const int slot = tid + i * 256;
    const int r = slot >> 5;
    const int c4 = (slot & 31) << 2;
    const int g = r0 + r;
    float4 t = make_float4(0.f, 0.f, 0.f, 0.f);
    if (g < N_NODES) t = *(const float4*)&h[(long long)g * DIM + c4];
    *(float4*)&T[r * TS + c4] = t;
  }
  __syncthreads();

  const int wave = tid >> 5;
  const int lane = tid & 31;
  const int m = lane & 15;
  const int hi = lane >> 4;
  const int rowb = wave * 16;

  v8f acc[8];
#pragma unroll
  for (int nt = 0; nt < 8; ++nt) acc[nt] = (v8f){};
  wmma_strip_gemm(T, LW, rowb, m, hi, acc);

#pragma unroll
  for (int nt = 0; nt < 8; ++nt) {
    const int c = nt * 16 + m;
    const float bias = bh[c];
#pragma unroll
    for (int v = 0; v < 8; ++v) {
      const int g = r0 + rowb + v + 8 * hi;
      if (g < N_NODES) out[(long long)g * DIM + c] = acc[nt][v] + bias;
    }
  }
}

// ---------------------------------------------------------------------------
// Driver
// ---------------------------------------------------------------------------
extern "C" void kernel_launch(void* const* d_in, const int* in_sizes, int n_in,
                              void* d_out, int out_size, void* d_ws, size_t ws_size,
                              hipStream_t stream) {
  (void)in_sizes; (void)n_in; (void)out_size; (void)ws_size;

  const float* x = (const float*)d_in[0];
  const long long* ei = (const long long*)d_in[1];   // int64 per reference
  const long long* src = ei;              // edge_index[0]
  const long long* dst = ei + N_EDGES;    // edge_index[1]

  const float* w1[3] = {(const float*)d_in[2], (const float*)d_in[6],  (const float*)d_in[10]};
  const float* b1[3] = {(const float*)d_in[3], (const float*)d_in[7],  (const float*)d_in[11]};
  const float* w2[3] = {(const float*)d_in[4], (const float*)d_in[8],  (const float*)d_in[12]};
  const float* b2[3] = {(const float*)d_in[5], (const float*)d_in[9],  (const float*)d_in[13]};
  const float* wh = (const float*)d_in[14];
  const float* bh = (const float*)d_in[15];

  // workspace: hbuf0 | hbuf1 | neigh  (3 * 25.6 MB)
  const size_t A = (size_t)N_NODES * DIM;
  float* ws = (float*)d_ws;
  float* hbuf0 = ws;
  float* hbuf1 = ws + A;
  float* neigh = ws + 2 * A;

  const dim3 blk(256);
  const int gemmGrid = (N_NODES + 127) / 128;       // 391 tiles
  const int scatGrid = (N_EDGES + 7) / 8;           // 8 edges/block (1/wave)
  const size_t smem3 = (size_t)3 * DIM * TS * sizeof(float);  // 202752 B
  const size_t smem2 = (size_t)2 * DIM * TS * sizeof(float);  // 135168 B

  const float* hin = x;
  float* houts[3] = {hbuf0, hbuf1, hbuf0};
  for (int l = 0; l < 3; ++l) {
    zero_f32<<<4096, blk, 0, stream>>>(neigh, (int)A);
    scatter_add<<<scatGrid, blk, 0, stream>>>(hin, src, dst, neigh);
    gin_layer<<<gemmGrid, blk, smem3, stream>>>(hin, neigh, w1[l], b1[l],
                                                w2[l], b2[l], houts[l],
                                                (l < 2) ? 1 : 0);
    hin = houts[l];
  }
  head_kernel<<<gemmGrid, blk, smem2, stream>>>(hin, wh, bh, (float*)d_out);
}